// UnrolledHH_15204184228603
// MI455X (gfx1250) — compile-verified
//
#include <hip/hip_runtime.h>
#include <cstdint>
#include <cstddef>

// ---------------- problem geometry ----------------
#define T_LEN 16384          // time steps per neuron (reference T)
#define NB    32             // neurons per block == one wave32 per workgroup
#define CHUNK 128            // time steps staged per LDS tile
#define PITCH 132            // LDS row pitch in floats: 132*4=528B, 16B-aligned rows,
                             // stride mod 64 == 4 -> only 2-way bank conflict on ds reads

typedef float v4f __attribute__((ext_vector_type(4)));

// ---------------- CDNA5 async global->LDS copy (ASYNCcnt) ----------------
__device__ __forceinline__ void async_load_b128(const float* g, float* l) {
    // low 32 bits of a flat shared pointer are the wave-relative LDS byte address
    unsigned lds_addr = (unsigned)(uintptr_t)l;
    unsigned long long gaddr = (unsigned long long)(uintptr_t)g;
    asm volatile("global_load_async_to_lds_b128 %0, %1, off"
                 :: "v"(lds_addr), "v"(gaddr)
                 : "memory");
}

__device__ __forceinline__ void wait_async0() {
    asm volatile("s_wait_asynccnt 0" ::: "memory");
}

// ---------------- HH rate functions ----------------
// exp factoring: exp(-(V+40)/10) = e^{1.5} * e55, exp(-(V+35)/10) = e^{2} * e55,
// with e55 = exp(-(V+55)/10). 4 TRANS ops/step instead of 6.
// v_rcp_f32 (~1 ulp) replaces the three IEEE divides on the loop-carried path.
__device__ __forceinline__ void hh_rates(float V, float& am, float& bm,
                                         float& ah, float& bh,
                                         float& an, float& bn) {
    const float EPS = 1e-8f;

    float x55 = V + 55.0f;
    float e55 = __expf(x55 * -0.1f);                   // exp(-(V+55)/10), v_exp_f32

    float x40 = V + 40.0f;
    float e40 = e55 * 4.4816890703380645f;             // exp(-(V+40)/10)
    float am_ = 0.1f * x40 * __builtin_amdgcn_rcpf(1.0f - e40 + EPS);
    am = (fabsf(x40) < 1e-6f) ? 1.0f : am_;

    bm = 4.0f  * __expf((V + 65.0f) * (-1.0f / 18.0f));
    ah = 0.07f * __expf((V + 65.0f) * -0.05f);

    float ebh = e55 * 7.3890560989306495f;             // exp(-(V+35)/10)
    bh = __builtin_amdgcn_rcpf(1.0f + ebh);

    float an_ = 0.01f * x55 * __builtin_amdgcn_rcpf(1.0f - e55 + EPS);
    an = (fabsf(x55) < 1e-6f) ? 0.1f : an_;

    bn = 0.125f * __expf((V + 65.0f) * -0.0125f);
}

__device__ __forceinline__ void hh_step(float& V, float& m, float& h, float& n,
                                        float It, float gNa, float gK, float gL) {
    // V_{t+1} depends only on current state (cheap chain); rates feed the gate
    // recurrences that close into V_{t+2}. Written V-first to aid pipelining.
    float m3   = m * m * m;
    float n2   = n * n;
    float n4   = n2 * n2;
    float I_Na = gNa * m3 * h * (V - 50.0f);     // E_NA =  50
    float I_K  = gK * n4 * (V + 77.0f);          // E_K  = -77
    float I_L  = gL * (V + 54.387f);             // E_L  = -54.387
    float Iion = I_Na + I_K + I_L;
    float Vn   = V + 0.02f * (It - Iion);        // DT = 0.02, C_M = 1

    float am, bm, ah, bh, an, bn;
    hh_rates(V, am, bm, ah, bh, an, bn);
    m = m + 0.02f * (am * (1.0f - m) - bm * m);
    h = h + 0.02f * (ah * (1.0f - h) - bh * h);
    n = n + 0.02f * (an * (1.0f - n) - bn * n);
    V = Vn;
}

// stable softplus, identical to jax.nn.softplus = max(x,0) + log1p(exp(-|x|))
__device__ __forceinline__ float softplusf(float x) {
    return fmaxf(x, 0.0f) + log1pf(__expf(-fabsf(x)));
}

// stage one [NB rows x CHUNK cols] tile: each lane copies 16B (4 consecutive
// timesteps) per row -> 32 lanes cover a full row (512B contiguous per wave)
__device__ __forceinline__ void stage_chunk(float (*buf)[PITCH], const float* gin,
                                            int t0, int tid, int nrows) {
    const float* gp = gin + t0 + tid * 4;
    #pragma unroll 8
    for (int r = 0; r < NB; ++r) {
        if (r < nrows)
            async_load_b128(gp + (size_t)r * T_LEN, &buf[r][tid * 4]);
    }
}

__global__ void __launch_bounds__(NB)
hh_scan_kernel(const float* __restrict__ I,
               const float* __restrict__ rNa,
               const float* __restrict__ rK,
               const float* __restrict__ rL,
               float* __restrict__ out, int B)
{
    __shared__ float sm[2][NB][PITCH];   // double-buffered input tiles (33 KB)

    const int tid   = threadIdx.x;                 // 0..31, one neuron per lane
    const int b0    = blockIdx.x * NB;
    const int b     = b0 + tid;
    int nrows       = B - b0;
    if (nrows > NB) nrows = NB;
    const bool wr   = (tid < nrows);

    const float gNa = softplusf(rNa[0]);
    const float gK  = softplusf(rK[0]);
    const float gL  = softplusf(rL[0]);

    // initial state at V0 = -65 (same construction as reference; exact divides, once)
    float am, bm, ah, bh, an, bn;
    hh_rates(-65.0f, am, bm, ah, bh, an, bn);
    float V = -65.0f;
    float m = am / (am + bm);
    float h = ah / (ah + bh);
    float n = an / (an + bn);

    const float* gin  = I + (size_t)b0 * T_LEN;            // block's input rows
    float*       orow = out + (size_t)b * (size_t)T_LEN;   // this lane's output row

    // prologue: stage chunk 0
    stage_chunk(sm[0], gin, 0, tid, nrows);

    const int NCH = T_LEN / CHUNK;                 // 128 chunks
    for (int k = 0; k < NCH; ++k) {
        wait_async0();                             // chunk k resident in LDS
        asm volatile("" ::: "memory");
        if (k + 1 < NCH)                           // prefetch next chunk (overlaps compute)
            stage_chunk(sm[(k + 1) & 1], gin, (k + 1) * CHUNK, tid, nrows);

        const float* row = sm[k & 1][tid];
        const int tbase  = k * CHUNK;
        v4f vout = {};
        #pragma unroll 8
        for (int c = 0; c < CHUNK; ++c) {
            float It = row[c];                     // merged into ds_load_b128 by 4
            vout[c & 3] = V;                       // record V BEFORE step:
                                                   // out[t] = V after consuming I[0..t-1]
            hh_step(V, m, h, n, It, gNa, gK, gL);
            if ((c & 3) == 3 && wr) {              // streaming NT store, 16B aligned
                __builtin_nontemporal_store(vout, (v4f*)(orow + tbase + c - 3));
            }
        }
    }
}

extern "C" void kernel_launch(void* const* d_in, const int* in_sizes, int n_in,
                              void* d_out, int out_size, void* d_ws, size_t ws_size,
                              hipStream_t stream) {
    const float* I   = (const float*)d_in[0];
    const float* rNa = (const float*)d_in[1];
    const float* rK  = (const float*)d_in[2];
    const float* rL  = (const float*)d_in[3];
    float* out = (float*)d_out;

    int B = in_sizes[0] / T_LEN;                   // 2048 for the reference shapes
    int grid = (B + NB - 1) / NB;                  // 64 single-wave workgroups
    hipLaunchKernelGGL(hh_scan_kernel, dim3(grid), dim3(NB), 0, stream,
                       I, rNa, rK, rL, out, B);
}